// RGCN_59485297050025
// MI455X (gfx1250) — compile-verified
//
#include <hip/hip_runtime.h>

// ---------------- problem constants (match reference) ----------------
static constexpr int DCH  = 128;           // channels
static constexpr int RREL = 5;             // relations
static constexpr int NLAY = 3;             // layers
static constexpr int KTOT = (RREL + 1) * DCH;   // 768 : 5 means + identity(h)
static constexpr int KPAD = KTOT + 4;      // LDS pad -> conflict-free banks

typedef __attribute__((ext_vector_type(2))) float v2f;
typedef __attribute__((ext_vector_type(8))) float v8f;

// ---------------- pass 0: pack B = [stack(W_r), root] into WMMA pair layout ---
// Bpack[(k/2)*DCH + col] = { B[k][col], B[k+1][col] }   (k even)
// boundary at k=640 is even -> a pair never straddles W/root.
__global__ __launch_bounds__(256) void rgcn_packB(const float* __restrict__ W,
                                                  const float* __restrict__ root,
                                                  float2* __restrict__ Bp) {
    int i = blockIdx.x * blockDim.x + threadIdx.x;      // over (KTOT/2)*DCH
    if (i >= (KTOT / 2) * DCH) return;
    int col = i & (DCH - 1);
    int k   = (i / DCH) * 2;
    const float* B0 = (k < RREL * DCH)
                        ? (W    + (size_t)k * DCH + col)
                        : (root + (size_t)(k - RREL * DCH) * DCH + col);
    Bp[i] = make_float2(B0[0], B0[DCH]);
}

// ---------------- pass 1: per-(dst,relation) edge counts ----------------
__global__ __launch_bounds__(256) void rgcn_count(const int* __restrict__ dst,
                                                  const int* __restrict__ et,
                                                  float* __restrict__ cnt, int nE) {
    int e = blockIdx.x * blockDim.x + threadIdx.x;
    if (e < nE) {
        unsafeAtomicAdd(&cnt[(size_t)dst[e] * RREL + et[e]], 1.0f);
    }
}

__global__ __launch_bounds__(256) void rgcn_invcnt(float* __restrict__ cnt, int n) {
    int i = blockIdx.x * blockDim.x + threadIdx.x;
    if (i < n) cnt[i] = 1.0f / fmaxf(cnt[i], 1.0f);
}

// ---------------- pass 2: scatter h[src] into sums[dst, rel, :] ----------------
// one wave32 per edge; each lane moves a float4 (32 lanes * 4 = 128 channels)
__global__ __launch_bounds__(256) void rgcn_scatter(const float* __restrict__ h,
                                                    const int* __restrict__ src,
                                                    const int* __restrict__ dst,
                                                    const int* __restrict__ et,
                                                    float* __restrict__ sums, int nE) {
    int wave = blockIdx.x * (blockDim.x >> 5) + (threadIdx.x >> 5);
    if (wave >= nE) return;
    int lane = threadIdx.x & 31;
    int s = src[wave];
    int d = dst[wave];
    int t = et[wave];
    const float4 v = *(const float4*)(h + (size_t)s * DCH + lane * 4);
    float* base = sums + ((size_t)d * RREL + t) * DCH + lane * 4;
    unsafeAtomicAdd(base + 0, v.x);
    unsafeAtomicAdd(base + 1, v.y);
    unsafeAtomicAdd(base + 2, v.z);
    unsafeAtomicAdd(base + 3, v.w);
}

// ---------------- pass 3: fused mean + GEMM + bias (+ReLU) via FP32 WMMA ------
// A = [16 nodes x 768]  (5 scaled relation-means + h), staged in LDS
// B = packed pairs from L2; one block = one 16-row tile; 8 waves = 8 col tiles
template <bool RELU>
__global__ __launch_bounds__(256) void rgcn_gemm(const float* __restrict__ sums,
                                                 const float* __restrict__ invcnt,
                                                 const float* __restrict__ h,
                                                 const float2* __restrict__ Bp,   // packed [KTOT/2][DCH]
                                                 const float* __restrict__ bias,  // [D] this layer
                                                 float* __restrict__ out,
                                                 int nNodes) {
    __shared__ float Alds[16][KPAD];

    const int tid   = threadIdx.x;
    const int node0 = blockIdx.x * 16;

    // ---- stage A tile into LDS (float4 granularity), scaling means by 1/cnt
    for (int g = tid; g < 16 * (KTOT / 4); g += 256) {
        int m    = g / (KTOT / 4);
        int k4   = (g % (KTOT / 4)) * 4;
        int node = node0 + m;
        if (node >= nNodes) node = nNodes - 1;   // clamp (N%16==0 in practice)
        int chunk = k4 >> 7;                     // k4 / 128
        int kd    = k4 & 127;
        float4 v;
        if (chunk < RREL) {
            v = *(const float4*)(sums + ((size_t)node * RREL + chunk) * DCH + kd);
            float sc = invcnt[(size_t)node * RREL + chunk];
            v.x *= sc; v.y *= sc; v.z *= sc; v.w *= sc;
        } else {
            v = *(const float4*)(h + (size_t)node * DCH + kd);
        }
        Alds[m][k4 + 0] = v.x;
        Alds[m][k4 + 1] = v.y;
        Alds[m][k4 + 2] = v.z;
        Alds[m][k4 + 3] = v.w;
    }
    __syncthreads();

    const int lane = tid & 31;
    const int half = lane >> 4;       // 0: K pair {k,k+1}, 1: {k+2,k+3}
    const int l16  = lane & 15;
    const int colT = (tid >> 5);      // wave id -> column tile
    const int col  = colT * 16 + l16; // output / B column

    // per-lane B stream: pair index (k/2 + half), linear stride of 2 pairs/iter
    const float2* bptr = Bp + (size_t)half * DCH + col;

    v8f c = {0.f, 0.f, 0.f, 0.f, 0.f, 0.f, 0.f, 0.f};

    #pragma unroll 8
    for (int k = 0; k < KTOT; k += 4) {
        const int ka = k + 2 * half;
        v2f a;
        a.x = Alds[l16][ka];                  // pairs merge -> ds_load_2addr_b64
        a.y = Alds[l16][ka + 1];
        const float2 bb = bptr[(size_t)(k >> 1) * DCH];  // one global_load_b64
        v2f b;
        b.x = bb.x;
        b.y = bb.y;
        // D = A(16x4) * B(4x16) + C  -> v_wmma_f32_16x16x4_f32
        c = __builtin_amdgcn_wmma_f32_16x16x4_f32(false, a, false, b,
                                                  (short)0, c, false, false);
    }

    // ---- epilogue: bias, relu, store per WMMA C/D layout
    const float bcol = bias[col];
    float r[8];
    #pragma unroll
    for (int v = 0; v < 8; ++v) {
        r[v] = c[v] + bcol;
        if (RELU) r[v] = fmaxf(r[v], 0.0f);
    }

    if (node0 + 16 <= nNodes) {
        // fast path: single base, immediate offsets (8 rows * 512B stride)
        float* obase = out + (size_t)(node0 + 8 * half) * DCH + col;
        #pragma unroll
        for (int v = 0; v < 8; ++v) obase[(size_t)v * DCH] = r[v];
    } else {
        #pragma unroll
        for (int v = 0; v < 8; ++v) {
            int row = node0 + v + 8 * half;
            if (row < nNodes) out[(size_t)row * DCH + col] = r[v];
        }
    }
}

// ---------------- driver ----------------
extern "C" void kernel_launch(void* const* d_in, const int* in_sizes, int n_in,
                              void* d_out, int out_size, void* d_ws, size_t ws_size,
                              hipStream_t stream) {
    const float* x       = (const float*)d_in[0];   // [N, D]
    const int*   eidx    = (const int*)  d_in[1];   // [2, E]
    const int*   etype   = (const int*)  d_in[2];   // [E]
    const float* weights = (const float*)d_in[3];   // [L, R, D, D]
    const float* roots   = (const float*)d_in[4];   // [L, D, D]
    const float* biases  = (const float*)d_in[5];   // [L, D]

    const int N = in_sizes[0] / DCH;
    const int E = in_sizes[2];
    const int* src = eidx;
    const int* dst = eidx + E;

    // workspace layout
    char*  ws     = (char*)d_ws;
    size_t szSums = (size_t)N * RREL * DCH * sizeof(float);   // 256 MB
    size_t szCnt  = (size_t)N * RREL * sizeof(float);         //   2 MB
    size_t szH    = (size_t)N * DCH * sizeof(float);          //  51 MB
    float*  sums = (float*)ws;
    float*  cnt  = (float*)(ws + szSums);
    float*  hA   = (float*)(ws + szSums + szCnt);
    float*  hB   = (float*)(ws + szSums + szCnt + szH);
    float2* Bp0  = (float2*)(ws + szSums + szCnt + 2 * szH);  // [L][KTOT/2][DCH]

    // one-time: pack per-layer B into WMMA pair layout
    {
        int nPk = (KTOT / 2) * DCH;
        for (int l = 0; l < NLAY; ++l) {
            rgcn_packB<<<(nPk + 255) / 256, 256, 0, stream>>>(
                weights + (size_t)l * RREL * DCH * DCH,
                roots   + (size_t)l * DCH * DCH,
                Bp0 + (size_t)l * (KTOT / 2) * DCH);
        }
    }

    // per-(node,relation) counts: edge-only, compute once, store reciprocal
    hipMemsetAsync(cnt, 0, szCnt, stream);
    rgcn_count<<<(E + 255) / 256, 256, 0, stream>>>(dst, etype, cnt, E);
    rgcn_invcnt<<<((N * RREL) + 255) / 256, 256, 0, stream>>>(cnt, N * RREL);

    const float* hin = x;
    for (int l = 0; l < NLAY; ++l) {
        hipMemsetAsync(sums, 0, szSums, stream);
        rgcn_scatter<<<(E + 7) / 8, 256, 0, stream>>>(hin, src, dst, etype, sums, E);
        float* hout = (l == 0) ? hA : (l == 1) ? hB : (float*)d_out;
        const float2* Bpl = Bp0 + (size_t)l * (KTOT / 2) * DCH;
        const float*  bl  = biases + (size_t)l * DCH;
        if (l < NLAY - 1) {
            rgcn_gemm<true><<<(N + 15) / 16, 256, 0, stream>>>(
                sums, cnt, hin, Bpl, bl, hout, N);
        } else {
            rgcn_gemm<false><<<(N + 15) / 16, 256, 0, stream>>>(
                sums, cnt, hin, Bpl, bl, hout, N);
        }
        hin = hout;
    }
}